// MaxSpanRepr_69844758168066
// MI455X (gfx1250) — compile-verified
//
#include <hip/hip_runtime.h>

// MaxSpanRepr: out[b,l,d] = max_s( mask[b,s] ? spans[b,l,s,d] : NEG_FILL )
// spans: [2048, 13, 4, 1024] f32, mask: [2048, 4] i32, out: [2048, 13, 1024] f32.
//
// HBM-bandwidth-bound streaming reduction. MI455X strategy:
//  - 2D grid (l, b): no integer division anywhere; block = 256 threads
//    (8 x wave32) covering D=1024 as float4
//  - mask row is block-uniform -> one s_load_b128 + scalar branches that
//    skip the global_load_b128 for masked span rows entirely (~halves
//    expected read traffic: ~327 MB total -> ~14 us at 23.3 TB/s)
//  - non-temporal (TH_NT) B128 loads/stores: data touched exactly once,
//    keep it out of WGP$/L2
//  - 32-bit element offsets (436 MB tensor) keep saddr+voffset addressing

typedef __attribute__((ext_vector_type(4))) float v4f;

#define NEG_FILL (-10000000000.0f)

static __device__ __forceinline__ v4f vmax4(v4f a, v4f b) {
    v4f r;
    r.x = fmaxf(a.x, b.x);
    r.y = fmaxf(a.y, b.y);
    r.z = fmaxf(a.z, b.z);
    r.w = fmaxf(a.w, b.w);
    return r;
}

__global__ __launch_bounds__(256) void
MaxSpanRepr_kernel(const float* __restrict__ spans,
                   const int*   __restrict__ amask,
                   float*       __restrict__ out)
{
    const unsigned l  = blockIdx.x;                     // layer   (uniform)
    const unsigned b  = blockIdx.y;                     // batch   (uniform)
    const unsigned bl = b * gridDim.x + l;              // (b, l) flat index
    const unsigned d4 = threadIdx.x;                    // float4 index in D=1024

    // Block-uniform mask row -> single s_load_b128, branches are scalar.
    const int* __restrict__ m = amask + b * 4u;
    const int m0 = m[0];
    const int m1 = m[1];
    const int m2 = m[2];
    const int m3 = m[3];

    // spans in v4f units: (b,l) group = 4 rows * 256 v4f; span s at +s*256.
    const v4f* __restrict__ p = (const v4f*)spans + (unsigned)(bl * 1024u) + d4;

    v4f acc = { NEG_FILL, NEG_FILL, NEG_FILL, NEG_FILL };

    // Uniform scalar branches: masked span rows are never loaded at all.
    if (m0) acc = vmax4(acc, __builtin_nontemporal_load(p + 0u));
    if (m1) acc = vmax4(acc, __builtin_nontemporal_load(p + 256u));
    if (m2) acc = vmax4(acc, __builtin_nontemporal_load(p + 512u));
    if (m3) acc = vmax4(acc, __builtin_nontemporal_load(p + 768u));

    __builtin_nontemporal_store(acc, (v4f*)out + (unsigned)(bl * 256u) + d4);
}

extern "C" void kernel_launch(void* const* d_in, const int* in_sizes, int n_in,
                              void* d_out, int out_size, void* d_ws, size_t ws_size,
                              hipStream_t stream) {
    const float* spans = (const float*)d_in[0];
    const int*   amask = (const int*)d_in[1];
    float*       out   = (float*)d_out;

    const int D      = 1024;
    const int S      = 4;
    const int B      = in_sizes[1] / S;      // 2048
    const int nBL    = out_size / D;         // 2048 * 13 = 26624
    const int layers = nBL / B;              // 13

    dim3 grid((unsigned)layers, (unsigned)B);   // (l, b): no division in kernel
    dim3 block((unsigned)(D / 4));              // 256 threads = 8 x wave32

    hipLaunchKernelGGL(MaxSpanRepr_kernel, grid, block, 0, stream,
                       spans, amask, out);
}